// GraphSAGEEncoder_13898514170445
// MI455X (gfx1250) — compile-verified
//
#include <hip/hip_runtime.h>

// GraphSAGE encoder for MI455X (gfx1250, wave32).
// Constants from the reference problem.
#define DD      128   // hidden dim
#define NCLUST  32    // clusters per graph
#define LSTR    132   // LDS row stride (pad 4 floats -> conflict-free column reads)

typedef __attribute__((ext_vector_type(2))) float v2f;
typedef __attribute__((ext_vector_type(8))) float v8f;

// ---------------- utility kernels ----------------

__global__ void zero_f4_kernel(float4* __restrict__ p, int n4) {
  int t = blockIdx.x * blockDim.x + threadIdx.x;
  if (t < n4) p[t] = make_float4(0.f, 0.f, 0.f, 0.f);
}

__global__ void zero_i_kernel(int* __restrict__ p, int n) {
  int t = blockIdx.x * blockDim.x + threadIdx.x;
  if (t < n) p[t] = 0;
}

// in-degree counts (layer-invariant; computed once)
__global__ void count_kernel(const int* __restrict__ dst, int* __restrict__ degc, int ne) {
  int t = blockIdx.x * blockDim.x + threadIdx.x;
  if (t < ne) atomicAdd(&degc[dst[t]], 1);
}

// x[i] = embed_table[deg_idx[i]] ; one float4 per thread (32 float4 per row)
__global__ void embed_kernel(const int* __restrict__ deg, const float4* __restrict__ tab,
                             float4* __restrict__ x, int n) {
  int t = blockIdx.x * blockDim.x + threadIdx.x;
  int i = t >> 5, c = t & 31;
  if (i < n) x[(size_t)i * 32 + c] = tab[(size_t)deg[i] * 32 + c];
}

// agg[dst] += x[src] ; one wave per edge, float4 per lane, fp32 atomics (L2-resident)
__global__ void scatter_kernel(const int* __restrict__ src, const int* __restrict__ dst,
                               const float4* __restrict__ x, float* __restrict__ agg, int ne) {
  int t = blockIdx.x * blockDim.x + threadIdx.x;
  int e = t >> 5, lane = t & 31;
  if (e >= ne) return;
  float4 v = x[(size_t)src[e] * 32 + lane];
  float* ap = agg + (size_t)dst[e] * DD + lane * 4;
  atomicAdd(ap + 0, v.x);
  atomicAdd(ap + 1, v.y);
  atomicAdd(ap + 2, v.z);
  atomicAdd(ap + 3, v.w);
}

// ---------------- fused SAGE layer: y = relu( mean(agg) @ Wl + bl + x @ Wr ) ----------------
// Block = 256 threads = 8 waves. Block computes rows [row0, row0+16) x all 128 cols.
// Wave w computes the 16x16 output tile at column offset 16*w via V_WMMA_F32_16X16X4_F32,
// accumulating both GEMMs into one 8-VGPR accumulator. Requires N % 16 == 0 (50000 = 3125*16).
__global__ __launch_bounds__(256) void sage_wmma_kernel(
    const float* __restrict__ agg, const int* __restrict__ degc,
    const float* __restrict__ x,
    const float* __restrict__ Wl, const float* __restrict__ bl,
    const float* __restrict__ Wr, float* __restrict__ y) {
  __shared__ float Am[16 * LSTR];  // mean-aggregated neighbor features
  __shared__ float Xm[16 * LSTR];  // self features

  const int row0 = blockIdx.x * 16;
  const int t = threadIdx.x;

  // Cooperative load of 16x128 tiles (2 x 2048 floats); fold mean into Am.
  for (int f = t; f < 512; f += 256) {
    const int r = f >> 5, c4 = f & 31;
    const int gr = row0 + r;
    const float4 a4 = reinterpret_cast<const float4*>(agg + (size_t)gr * DD)[c4];
    const float4 x4 = reinterpret_cast<const float4*>(x + (size_t)gr * DD)[c4];
    const int dc = degc[gr];
    const float inv = 1.0f / (float)(dc > 0 ? dc : 1);
    float* am = &Am[r * LSTR + c4 * 4];
    float* xm = &Xm[r * LSTR + c4 * 4];
    am[0] = a4.x * inv; am[1] = a4.y * inv; am[2] = a4.z * inv; am[3] = a4.w * inv;
    xm[0] = x4.x;       xm[1] = x4.y;       xm[2] = x4.z;       xm[3] = x4.w;
  }
  __syncthreads();

  const int lane = t & 31;
  const int wv   = t >> 5;            // 0..7 -> output column tile
  const int n0   = wv * 16;
  const int m    = lane & 15;         // A row / B,D column within tile
  const int kh   = (lane >> 4) << 1;  // lanes 0-15: K=0,1 ; lanes 16-31: K=2,3

  v8f c = {};

#pragma unroll
  for (int kb = 0; kb < 32; ++kb) {   // K = 128, 4 per WMMA
    const int k = (kb << 2) + kh;
    v2f a, b;
    a.x = Am[m * LSTR + k];
    a.y = Am[m * LSTR + k + 1];
    b.x = Wl[k * DD + n0 + m];
    b.y = Wl[(k + 1) * DD + n0 + m];
    c = __builtin_amdgcn_wmma_f32_16x16x4_f32(false, a, false, b, (short)0, c, false, false);
  }
#pragma unroll
  for (int kb = 0; kb < 32; ++kb) {
    const int k = (kb << 2) + kh;
    v2f a, b;
    a.x = Xm[m * LSTR + k];
    a.y = Xm[m * LSTR + k + 1];
    b.x = Wr[k * DD + n0 + m];
    b.y = Wr[(k + 1) * DD + n0 + m];
    c = __builtin_amdgcn_wmma_f32_16x16x4_f32(false, a, false, b, (short)0, c, false, false);
  }

  // C/D layout: lane holds column n = lane%16; rows r (+8 for lanes 16-31).
  const float bias = bl[n0 + m];
  const int mb = (lane >> 4) << 3;  // 0 or 8
#pragma unroll
  for (int r = 0; r < 8; ++r) {
    float v = c[r] + bias;
    y[(size_t)(row0 + mb + r) * DD + n0 + m] = v > 0.f ? v : 0.f;
  }
}

// ---------------- pooling ----------------

__global__ void pool_scatter_kernel(const float4* __restrict__ x, const int* __restrict__ batch,
                                    const int* __restrict__ parts, float* __restrict__ out,
                                    int* __restrict__ poolc, int n) {
  int t = blockIdx.x * blockDim.x + threadIdx.x;
  int i = t >> 5, lane = t & 31;
  if (i >= n) return;
  int s = batch[i] * NCLUST + parts[i];
  float4 v = x[(size_t)i * 32 + lane];
  float* op = out + (size_t)s * DD + lane * 4;
  atomicAdd(op + 0, v.x);
  atomicAdd(op + 1, v.y);
  atomicAdd(op + 2, v.z);
  atomicAdd(op + 3, v.w);
  if (lane == 0) atomicAdd(&poolc[s], 1);
}

__global__ void pool_div_kernel(float* __restrict__ out, const int* __restrict__ poolc, int total) {
  int t = blockIdx.x * blockDim.x + threadIdx.x;
  if (t < total) {
    int cnt = poolc[t >> 7];  // t / DD
    out[t] = cnt > 0 ? out[t] / (float)cnt : 0.0f;
  }
}

// ---------------- host orchestration ----------------

extern "C" void kernel_launch(void* const* d_in, const int* in_sizes, int n_in,
                              void* d_out, int out_size, void* d_ws, size_t ws_size,
                              hipStream_t stream) {
  const int*   deg_idx = (const int*)d_in[0];
  const int*   ei      = (const int*)d_in[1];
  const int*   batch   = (const int*)d_in[2];
  const int*   parts   = (const int*)d_in[3];
  const float* table   = (const float*)d_in[4];
  const float* Wl      = (const float*)d_in[5];
  const float* bl      = (const float*)d_in[6];
  const float* Wr      = (const float*)d_in[7];
  float* out = (float*)d_out;

  const int n  = in_sizes[0];       // 50000 (multiple of 16)
  const int ne = in_sizes[1] / 2;   // 800000
  const int* src = ei;
  const int* dst = ei + ne;

  // workspace: x | y | agg | degc | poolc
  float* x   = (float*)d_ws;
  float* yb  = x + (size_t)n * DD;
  float* agg = yb + (size_t)n * DD;
  int* degc  = (int*)(agg + (size_t)n * DD);
  int* poolc = degc + n;

  const int B = 256;
  auto blks = [](long long v, int b) { return (int)((v + b - 1) / b); };

  zero_i_kernel<<<blks(n, B), B, 0, stream>>>(degc, n);
  zero_i_kernel<<<blks(512, B), B, 0, stream>>>(poolc, 512);
  zero_f4_kernel<<<blks(out_size / 4, B), B, 0, stream>>>((float4*)out, out_size / 4);
  count_kernel<<<blks(ne, B), B, 0, stream>>>(dst, degc, ne);
  embed_kernel<<<blks((long long)n * 32, B), B, 0, stream>>>(deg_idx, (const float4*)table,
                                                             (float4*)x, n);

  float* cur = x;
  float* nxt = yb;
  for (int l = 0; l < 3; ++l) {
    zero_f4_kernel<<<blks((long long)n * 32, B), B, 0, stream>>>((float4*)agg, n * 32);
    scatter_kernel<<<blks((long long)ne * 32, B), B, 0, stream>>>(src, dst, (const float4*)cur,
                                                                  agg, ne);
    sage_wmma_kernel<<<n / 16, B, 0, stream>>>(agg, degc, cur, Wl + (size_t)l * DD * DD,
                                               bl + (size_t)l * DD, Wr + (size_t)l * DD * DD, nxt);
    float* tmp = cur; cur = nxt; nxt = tmp;
  }

  pool_scatter_kernel<<<blks((long long)n * 32, B), B, 0, stream>>>((const float4*)cur, batch,
                                                                    parts, out, poolc, n);
  pool_div_kernel<<<blks(out_size, B), B, 0, stream>>>(out, poolc, out_size);
}